// IGN_2to1_Dense_18580028522835
// MI455X (gfx1250) — compile-verified
//
#include <hip/hip_runtime.h>

typedef __attribute__((ext_vector_type(2))) float v2f;
typedef __attribute__((ext_vector_type(8))) float v8f;

#define N_ 16
#define M_ 256
#define D_ 128
#define S_ 128
// stat layout per n: [3][D][m] = diag, rows, cols  (k = sec*128+d, contiguous B matrix [384 x 256])
#define STAT_PER_N (3 * D_ * M_)           // 98304 floats
#define STAT_TOTAL (N_ * STAT_PER_N)       // 1572864 floats
#define SD_OFF     STAT_TOTAL              // [N,D]
#define TOT_OFF    (SD_OFF + N_ * D_)      // [N,D]
#define CONST_OFF  (TOT_OFF + N_ * D_)     // [N,S]
#define AMAT_OFF   (CONST_OFF + N_ * S_)   // [384,128] k-major
#define GATE_OFF   (AMAT_OFF + 384 * 128)  // [N,m]
#define WS_FLOATS  (GATE_OFF + N_ * M_)

// ---- K0: zero the stat accumulator region ----
__global__ void k_zero(float4* __restrict__ p, int n4) {
    int idx = blockIdx.x * 256 + threadIdx.x;
    if (idx < n4) p[idx] = make_float4(0.f, 0.f, 0.f, 0.f);
}

// ---- K1: single streaming pass: diag + row/col sums ----
// grid = N*16 blocks (each owns n, i-tile of 16 rows), 256 threads.
// thread t: q = t&31 -> d4 = 4*q (float4 over d), jg = t>>5 -> 32 j's.
__global__ void k_stream(const float* __restrict__ in, float* __restrict__ stat) {
    const int bid = blockIdx.x;
    const int n  = bid >> 4;
    const int i0 = (bid & 15) << 4;
    const int t  = threadIdx.x;
    const int d4 = (t & 31) << 2;
    const int jg = t >> 5;

    float4 racc[16];
#pragma unroll
    for (int ii = 0; ii < 16; ++ii) racc[ii] = make_float4(0.f, 0.f, 0.f, 0.f);

    const float* base_n = in + (size_t)n * (M_ * M_ * D_);
    float* diagp = stat + (size_t)(n * 3 + 0) * D_ * M_;
    float* rowp  = stat + (size_t)(n * 3 + 1) * D_ * M_;
    float* colp  = stat + (size_t)(n * 3 + 2) * D_ * M_;

    for (int jj = 0; jj < 32; ++jj) {
        const int j = (jg << 5) + jj;
        const int di = j - i0;  // which ii (if any) hits the diagonal
        float4 cacc = make_float4(0.f, 0.f, 0.f, 0.f);
        const float* p = base_n + ((size_t)i0 * M_ + j) * D_ + d4;
#pragma unroll
        for (int ii = 0; ii < 16; ++ii) {
            float4 v = *(const float4*)(p + (size_t)ii * (M_ * D_));
            cacc.x += v.x; cacc.y += v.y; cacc.z += v.z; cacc.w += v.w;
            racc[ii].x += v.x; racc[ii].y += v.y; racc[ii].z += v.z; racc[ii].w += v.w;
            if (ii == di) {  // diagonal element inputs[n, j, j, d]
                diagp[(d4 + 0) * M_ + j] = v.x;
                diagp[(d4 + 1) * M_ + j] = v.y;
                diagp[(d4 + 2) * M_ + j] = v.z;
                diagp[(d4 + 3) * M_ + j] = v.w;
            }
        }
        atomicAdd(&colp[(d4 + 0) * M_ + j], cacc.x);
        atomicAdd(&colp[(d4 + 1) * M_ + j], cacc.y);
        atomicAdd(&colp[(d4 + 2) * M_ + j], cacc.z);
        atomicAdd(&colp[(d4 + 3) * M_ + j], cacc.w);
    }
#pragma unroll
    for (int ii = 0; ii < 16; ++ii) {
        atomicAdd(&rowp[(d4 + 0) * M_ + i0 + ii], racc[ii].x);
        atomicAdd(&rowp[(d4 + 1) * M_ + i0 + ii], racc[ii].y);
        atomicAdd(&rowp[(d4 + 2) * M_ + i0 + ii], racc[ii].z);
        atomicAdd(&rowp[(d4 + 3) * M_ + i0 + ii], racc[ii].w);
    }
}

// ---- K2: sum_diag and tot per (n,d) ----  grid N, block D
__global__ void k_sdtot(const float* __restrict__ stat, float* __restrict__ sd, float* __restrict__ tot) {
    const int n = blockIdx.x, d = threadIdx.x;
    const float* diagp = stat + ((size_t)(n * 3 + 0) * D_ + d) * M_;
    const float* rowp  = stat + ((size_t)(n * 3 + 1) * D_ + d) * M_;
    float a = 0.f, b = 0.f;
    for (int i = 0; i < M_; ++i) { a += diagp[i]; b += rowp[i]; }
    sd[n * D_ + d] = a;
    tot[n * D_ + d] = b;
}

// ---- K3: i-independent terms (b=1,4) + bias -> constn[n,s] ----  grid N, block S
__global__ void k_const(const float* __restrict__ coeffs, const float* __restrict__ bias,
                        const float* __restrict__ sd, const float* __restrict__ tot,
                        float* __restrict__ constn) {
    const int n = blockIdx.x, s = threadIdx.x;
    const float inv_m = 1.f / (float)M_, invD = 1.f / (float)D_;
    float acc = 0.f;
    for (int d = 0; d < D_; ++d) {
        const float* c = coeffs + ((size_t)d * S_ + s) * 5;
        acc += c[1] * sd[n * D_ + d] * inv_m + c[4] * tot[n * D_ + d] * (inv_m * inv_m);
    }
    constn[n * S_ + s] = bias[s] + acc * invD;
}

// ---- K4: fold coeffs (b=0,2,3) with 1/D, 1/m scales -> Amat[k][s], k in [0,384) ----
__global__ void k_amat(const float* __restrict__ coeffs, float* __restrict__ Amat) {
    const int idx = blockIdx.x * 256 + threadIdx.x;
    if (idx >= 384 * S_) return;
    const int k = idx >> 7, s = idx & (S_ - 1);
    const int sec = k >> 7, d = k & (D_ - 1);
    const int b = (sec == 0) ? 0 : ((sec == 1) ? 2 : 3);
    const float scale = (1.f / (float)D_) * ((sec == 0) ? 1.f : (1.f / (float)M_));
    Amat[idx] = coeffs[((size_t)d * S_ + s) * 5 + b] * scale;
}

// ---- K5: gate[n,q] = (sum_p mask[n,p,q] > 0) ----  grid N, block M
__global__ void k_gate(const float* __restrict__ mask, float* __restrict__ gate) {
    const int n = blockIdx.x, q = threadIdx.x;
    const float* mp = mask + (size_t)n * M_ * M_ + q;
    float g = 0.f;
    for (int p = 0; p < M_; ++p) g += mp[(size_t)p * M_];
    gate[n * M_ + q] = (g > 0.f) ? 1.f : 0.f;
}

// ---- K6: WMMA f32 GEMM: out[n,i,s] = (Amat . stat_n)[s,i] + constn, gated ----
// 2048 tiles of 16x16; block = 8 waves (256 thr), each wave one tile; K=384 -> 96 v_wmma.
__global__ void k_gemm_wmma(const float* __restrict__ Amat, const float* __restrict__ stat,
                            const float* __restrict__ constn, const float* __restrict__ gate,
                            float* __restrict__ out) {
    const int wave = threadIdx.x >> 5;
    const int lane = threadIdx.x & 31;
    const int tile = blockIdx.x * 8 + wave;           // 0..2047, exact
    const int n   = tile >> 7;
    const int rem = tile & 127;
    const int s0  = (rem >> 4) << 4;                  // 0..112
    const int i0  = (rem & 15) << 4;                  // 0..240
    const int half = lane >> 4;                       // A/B K-half per ISA layout
    const int l    = lane & 15;

    const float* statn = stat + (size_t)n * STAT_PER_N;   // B matrix [384 x 256], row-major
    v8f acc = {};
    for (int k = 0; k < 384; k += 4) {
        const int kk = k + (half ? 2 : 0);
        v2f a, b;
        a.x = Amat[kk * S_ + s0 + l];
        a.y = Amat[(kk + 1) * S_ + s0 + l];
        b.x = statn[kk * M_ + i0 + l];
        b.y = statn[(kk + 1) * M_ + i0 + l];
        acc = __builtin_amdgcn_wmma_f32_16x16x4_f32(
            /*neg_a=*/false, a, /*neg_b=*/false, b,
            /*c_mod=*/(short)0, acc, /*reuse_a=*/false, /*reuse_b=*/false);
    }
    // C layout: VGPR v, lanes 0-15 -> M=v (s0+v), lanes 16-31 -> M=v+8; N = l (i0+l)
    const float g = gate[n * M_ + i0 + l];
    const int sbase = s0 + (half ? 8 : 0);
    float* op = out + ((size_t)(n * M_ + i0 + l)) * S_;
    const float* cn = constn + n * S_;
#pragma unroll
    for (int v = 0; v < 8; ++v) {
        const int s = sbase + v;
        op[s] = (acc[v] + cn[s]) * g;   // 8 contiguous floats per lane
    }
}

extern "C" void kernel_launch(void* const* d_in, const int* in_sizes, int n_in,
                              void* d_out, int out_size, void* d_ws, size_t ws_size,
                              hipStream_t stream) {
    const float* inputs = (const float*)d_in[0];
    const float* mask   = (const float*)d_in[1];
    const float* coeffs = (const float*)d_in[2];
    const float* bias   = (const float*)d_in[3];
    float* out = (float*)d_out;
    float* ws  = (float*)d_ws;

    float* stat   = ws;
    float* sd     = ws + SD_OFF;
    float* tot    = ws + TOT_OFF;
    float* constn = ws + CONST_OFF;
    float* Amat   = ws + AMAT_OFF;
    float* gate   = ws + GATE_OFF;

    // K0: zero row/col accumulators (whole stat region)
    {
        int n4 = STAT_TOTAL / 4;
        int blocks = (n4 + 255) / 256;
        k_zero<<<blocks, 256, 0, stream>>>((float4*)stat, n4);
    }
    // K1: the only heavy kernel — one pass over 512 MiB
    k_stream<<<N_ * 16, 256, 0, stream>>>(inputs, stat);
    // K2..K5: tiny prep kernels
    k_sdtot<<<N_, D_, 0, stream>>>(stat, sd, tot);
    k_const<<<N_, S_, 0, stream>>>(coeffs, bias, sd, tot, constn);
    k_amat<<<(384 * S_ + 255) / 256, 256, 0, stream>>>(coeffs, Amat);
    k_gate<<<N_, M_, 0, stream>>>(mask, gate);
    // K6: WMMA contraction + epilogue
    k_gemm_wmma<<<256, 256, 0, stream>>>(Amat, stat, constn, gate, out);
}